// Net_53807350284776
// MI455X (gfx1250) — compile-verified
//
#include <hip/hip_runtime.h>
#include <hip/hip_bf16.h>

typedef __attribute__((ext_vector_type(2))) float v2f;
typedef __attribute__((ext_vector_type(8))) float v8f;

// ---------------------------------------------------------------------------
// Row L1 norm: invn[row] = 1 / max(sum|x[row,:]|, 1e-12)
// ---------------------------------------------------------------------------
__global__ void l1norm_kernel(const float* __restrict__ x, float* __restrict__ invn,
                              int D)
{
    const int row = blockIdx.x;
    __shared__ float sh[128];
    const float* xr = x + (size_t)row * D;
    float s = 0.0f;
    for (int i = threadIdx.x; i < D; i += blockDim.x) s += fabsf(xr[i]);
    sh[threadIdx.x] = s;
    __syncthreads();
    for (int st = 64; st > 0; st >>= 1) {
        if (threadIdx.x < st) sh[threadIdx.x] += sh[threadIdx.x + st];
        __syncthreads();
    }
    if (threadIdx.x == 0) invn[row] = 1.0f / fmaxf(sh[0], 1e-12f);
}

// ---------------------------------------------------------------------------
// FP32 WMMA GEMM: C[:, colOff:colOff+Ncols] = diag(rowScale) * A @ W (+ bias)
// A: [M,K] row-major (lda), W: [K,Ncols] row-major (ldw), C: row-major (ldc).
// One wave computes one 16x16 tile via V_WMMA_F32_16X16X4_F32.
// M % 16 == 0, K % 4 == 0 guaranteed by caller.
// Out-of-range columns: loads are CLAMPED (branch-free; WMMA column n only
// depends on B in lanes mr==n, and those columns are masked at the store),
// stores are guarded.
// ---------------------------------------------------------------------------
__global__ void gemm_wmma_f32(const float* __restrict__ A, int lda,
                              const float* __restrict__ rowScale,
                              const float* __restrict__ W, int ldw,
                              const float* __restrict__ bias,
                              float* __restrict__ C, int ldc, int colOff,
                              int M, int K, int Ncols)
{
    const int lane  = threadIdx.x & 31;
    const int wave  = threadIdx.x >> 5;
    const int MT    = M >> 4;
    const int mtile = blockIdx.x * (blockDim.x >> 5) + wave;
    if (mtile >= MT) return;                      // uniform per wave
    const int ntile = blockIdx.y;

    const int hi  = lane >> 4;                    // 0: K pair {0,1}, 1: {2,3}
    const int mr  = lane & 15;
    const int rowA  = (mtile << 4) + mr;
    const int col   = (ntile << 4) + mr;
    const int colL  = (col < Ncols) ? col : (Ncols - 1);   // clamped, branch-free

    const float s = rowScale ? rowScale[rowA] : 1.0f;
    const float* arow = A + (size_t)rowA * lda + 2 * hi;
    const float* wcol = W + (size_t)(2 * hi) * ldw + colL;
    const size_t wstep = (size_t)4 * ldw;

    v8f acc = {0.f, 0.f, 0.f, 0.f, 0.f, 0.f, 0.f, 0.f};
    for (int k = 0; k < K; k += 4) {
        v2f a, b;
        a.x = arow[0] * s;
        a.y = arow[1] * s;
        b.x = wcol[0];
        b.y = wcol[ldw];
        acc = __builtin_amdgcn_wmma_f32_16x16x4_f32(
            /*neg_a=*/false, a, /*neg_b=*/false, b,
            /*c_mod=*/(short)0, acc, /*reuse_a=*/false, /*reuse_b=*/false);
        arow += 4;
        wcol += wstep;
    }

    if (col < Ncols) {
        const float bv = bias ? bias[col] : 0.0f;
        #pragma unroll
        for (int r = 0; r < 8; ++r) {
            const int m = (mtile << 4) + r + 8 * hi;
            C[(size_t)m * ldc + colOff + col] = acc[r] + bv;
        }
    }
}

// ---------------------------------------------------------------------------
// deg[dst[e]] += 1
// ---------------------------------------------------------------------------
__global__ void count_deg_kernel(const int* __restrict__ dst, float* __restrict__ deg, int E)
{
    const int e = blockIdx.x * blockDim.x + threadIdx.x;
    if (e < E) unsafeAtomicAdd(&deg[dst[e]], 1.0f);
}

__global__ void invert_deg_kernel(float* __restrict__ deg, int N)
{
    const int n = blockIdx.x * blockDim.x + threadIdx.x;
    if (n < N) deg[n] = 1.0f / fmaxf(deg[n], 1.0f);
}

// ---------------------------------------------------------------------------
// Edge scatter: agg[dst, c] += h[src, c], c in [0,64).
// 256 threads/block = 16 edges/block, 16 threads/edge, float4 per thread.
// Edge indices staged in LDS once per block (kills 16x redundant index loads).
// ---------------------------------------------------------------------------
__global__ void edge_agg_kernel(const float* __restrict__ h, int ldh,
                                float* __restrict__ agg,
                                const int* __restrict__ src,
                                const int* __restrict__ dst,
                                int E)
{
    __shared__ int ssrc[16];
    __shared__ int sdst[16];
    const int e0 = blockIdx.x << 4;
    if (threadIdx.x < 16) {
        const int e = e0 + threadIdx.x;
        ssrc[threadIdx.x] = (e < E) ? src[e] : -1;
        sdst[threadIdx.x] = (e < E) ? dst[e] : 0;
    }
    __syncthreads();
    const int li = threadIdx.x >> 4;          // local edge 0..15
    const int cg = (threadIdx.x & 15) << 2;   // channel group *4
    const int s = ssrc[li];
    if (s < 0) return;
    const int d = sdst[li];
    const float4 v = *(const float4*)(h + (size_t)s * ldh + cg);
    float* ad = agg + (size_t)d * 64 + cg;
    unsafeAtomicAdd(ad + 0, v.x);
    unsafeAtomicAdd(ad + 1, v.y);
    unsafeAtomicAdd(ad + 2, v.z);
    unsafeAtomicAdd(ad + 3, v.w);
}

// ---------------------------------------------------------------------------
// e[n,c] = agg[n,c] * deginv[n] + bl[c] + h_r[n,c]   (float4 per thread)
// ---------------------------------------------------------------------------
__global__ void combine_kernel(const float* __restrict__ agg,
                               const float* __restrict__ deginv,
                               const float* __restrict__ bl,
                               const float* __restrict__ hr, int ldh,
                               float* __restrict__ eo, int total16)
{
    const int gid = blockIdx.x * blockDim.x + threadIdx.x;
    if (gid >= total16) return;
    const int n = gid >> 4, c = (gid & 15) << 2;
    const float di = deginv[n];
    const float4 a = *(const float4*)(agg + (size_t)n * 64 + c);
    const float4 r = *(const float4*)(hr + (size_t)n * ldh + c);
    const float4 b = *(const float4*)(bl + c);
    float4 o;
    o.x = a.x * di + b.x + r.x;
    o.y = a.y * di + b.y + r.y;
    o.z = a.z * di + b.z + r.z;
    o.w = a.w * di + b.w + r.w;
    *(float4*)(eo + (size_t)n * 64 + c) = o;
}

// ---------------------------------------------------------------------------
// Global mean pool (accumulate) + counts
// ---------------------------------------------------------------------------
__global__ void pool_kernel(const float* __restrict__ e, const int* __restrict__ batch,
                            float* __restrict__ pool, int total16)
{
    const int gid = blockIdx.x * blockDim.x + threadIdx.x;
    if (gid >= total16) return;
    const int n = gid >> 4, c = (gid & 15) << 2;
    const int b = batch[n];
    const float4 v = *(const float4*)(e + (size_t)n * 64 + c);
    float* pd = pool + (size_t)b * 64 + c;
    unsafeAtomicAdd(pd + 0, v.x);
    unsafeAtomicAdd(pd + 1, v.y);
    unsafeAtomicAdd(pd + 2, v.z);
    unsafeAtomicAdd(pd + 3, v.w);
}

__global__ void cnt_kernel(const int* __restrict__ batch, float* __restrict__ cnt, int N)
{
    const int n = blockIdx.x * blockDim.x + threadIdx.x;
    if (n < N) unsafeAtomicAdd(&cnt[batch[n]], 1.0f);
}

__global__ void pool_div_kernel(float* __restrict__ pool, const float* __restrict__ cnt, int total)
{
    const int gid = blockIdx.x * blockDim.x + threadIdx.x;
    if (gid >= total) return;
    pool[gid] *= 1.0f / fmaxf(cnt[gid >> 6], 1.0f);
}

// ---------------------------------------------------------------------------
// BatchNorm1d (training stats, biased var) + tanh, in place.
// One block per column; `rows` == 512.
// ---------------------------------------------------------------------------
__global__ void bn_tanh_kernel(float* __restrict__ T, int rows, int cols,
                               const float* __restrict__ g, const float* __restrict__ beta)
{
    const int j = blockIdx.x;
    __shared__ float ssum[256];
    __shared__ float ssq[256];
    float s = 0.0f, q = 0.0f;
    for (int i = threadIdx.x; i < rows; i += blockDim.x) {
        const float v = T[(size_t)i * cols + j];
        s += v; q += v * v;
    }
    ssum[threadIdx.x] = s; ssq[threadIdx.x] = q;
    __syncthreads();
    for (int st = blockDim.x >> 1; st > 0; st >>= 1) {
        if (threadIdx.x < st) {
            ssum[threadIdx.x] += ssum[threadIdx.x + st];
            ssq[threadIdx.x]  += ssq[threadIdx.x + st];
        }
        __syncthreads();
    }
    const float mu  = ssum[0] / (float)rows;
    const float var = ssq[0] / (float)rows - mu * mu;
    const float sc  = rsqrtf(var + 1e-5f) * g[j];
    const float bb  = beta[j];
    for (int i = threadIdx.x; i < rows; i += blockDim.x) {
        const float v = T[(size_t)i * cols + j];
        T[(size_t)i * cols + j] = tanhf((v - mu) * sc + bb);
    }
}

// ---------------------------------------------------------------------------
// Launch
// ---------------------------------------------------------------------------
extern "C" void kernel_launch(void* const* d_in, const int* in_sizes, int n_in,
                              void* d_out, int out_size, void* d_ws, size_t ws_size,
                              hipStream_t stream)
{
    const float* x      = (const float*)d_in[0];
    const int*   eidx   = (const int*)d_in[1];
    const int*   batch  = (const int*)d_in[2];
    // d_in[3] = y (unused)
    const float* c1Wl = (const float*)d_in[4];
    const float* c1bl = (const float*)d_in[5];
    const float* c1Wr = (const float*)d_in[6];
    const float* c2Wl = (const float*)d_in[7];
    const float* c2bl = (const float*)d_in[8];
    const float* c2Wr = (const float*)d_in[9];
    const float* c3Wl = (const float*)d_in[10];
    const float* c3bl = (const float*)d_in[11];
    const float* c3Wr = (const float*)d_in[12];
    const float* l1W = (const float*)d_in[13];
    const float* l1b = (const float*)d_in[14];
    const float* bn1g = (const float*)d_in[15];
    const float* bn1b = (const float*)d_in[16];
    const float* l2W = (const float*)d_in[17];
    const float* l2b = (const float*)d_in[18];
    const float* bn2g = (const float*)d_in[19];
    const float* bn2b = (const float*)d_in[20];
    const float* l3W = (const float*)d_in[21];
    const float* l3b = (const float*)d_in[22];
    const float* bn3g = (const float*)d_in[23];
    const float* bn3b = (const float*)d_in[24];
    const float* l4W = (const float*)d_in[25];
    const float* l4b = (const float*)d_in[26];
    float* out = (float*)d_out;

    const int N    = in_sizes[2];          // 100000
    const int E    = in_sizes[1] / 2;      // 3.2M
    const int DIN  = in_sizes[0] / N;      // 500
    const int DH   = in_sizes[5];          // 64
    const int B    = in_sizes[3];          // 512
    const int HID  = in_sizes[14];         // 200
    const int H2   = in_sizes[18];         // 100
    const int NCL  = in_sizes[26];         // 80

    const int* src = eidx;
    const int* dst = eidx + E;

    // Workspace layout (floats)
    float* ws     = (float*)d_ws;
    float* invn   = ws;                    // N
    float* deg    = invn + N;              // N  (becomes deginv)
    float* h      = deg + N;               // N * 128
    float* agg    = h + (size_t)N * 128;   // N * 64
    float* e      = agg + (size_t)N * 64;  // N * 64
    float* pool   = e + (size_t)N * 64;    // B * 64
    float* cnt    = pool + (size_t)B * 64; // B
    float* t1     = cnt + B;               // B * HID
    float* t2     = t1 + (size_t)B * HID;  // B * HID

    const int TPB = 256;
    // ---- normalization scale + degrees ----
    l1norm_kernel<<<N, 128, 0, stream>>>(x, invn, DIN);
    hipMemsetAsync(deg, 0, (size_t)N * sizeof(float), stream);
    count_deg_kernel<<<(E + TPB - 1) / TPB, TPB, 0, stream>>>(dst, deg, E);
    invert_deg_kernel<<<(N + TPB - 1) / TPB, TPB, 0, stream>>>(deg, N);

    const int eBlocks = (E + 15) / 16;          // 16 edges per 256-thread block
    const int nTot16 = N * 16;                  // float4 granularity over N x 64
    const int nBlocks16 = (nTot16 + TPB - 1) / TPB;

    // GEMM launch config: 4 waves (128 threads) per block
    const int GB = 128;
    const int MTn = N >> 4;
    dim3 gridConv((MTn + 3) / 4, (DH + 15) / 16);

    // ===== conv1: h_l = (invn*x)@Wl  (cols 0..63), h_r = (invn*x)@Wr (cols 64..127)
    gemm_wmma_f32<<<gridConv, GB, 0, stream>>>(x, DIN, invn, c1Wl, DH, nullptr, h, 128, 0,  N, DIN, DH);
    gemm_wmma_f32<<<gridConv, GB, 0, stream>>>(x, DIN, invn, c1Wr, DH, nullptr, h, 128, DH, N, DIN, DH);
    hipMemsetAsync(agg, 0, (size_t)N * 64 * sizeof(float), stream);
    edge_agg_kernel<<<eBlocks, TPB, 0, stream>>>(h, 128, agg, src, dst, E);
    combine_kernel<<<nBlocks16, TPB, 0, stream>>>(agg, deg, c1bl, h + DH, 128, e, nTot16);

    // ===== conv2
    gemm_wmma_f32<<<gridConv, GB, 0, stream>>>(e, DH, nullptr, c2Wl, DH, nullptr, h, 128, 0,  N, DH, DH);
    gemm_wmma_f32<<<gridConv, GB, 0, stream>>>(e, DH, nullptr, c2Wr, DH, nullptr, h, 128, DH, N, DH, DH);
    hipMemsetAsync(agg, 0, (size_t)N * 64 * sizeof(float), stream);
    edge_agg_kernel<<<eBlocks, TPB, 0, stream>>>(h, 128, agg, src, dst, E);
    combine_kernel<<<nBlocks16, TPB, 0, stream>>>(agg, deg, c2bl, h + DH, 128, e, nTot16);

    // ===== conv3
    gemm_wmma_f32<<<gridConv, GB, 0, stream>>>(e, DH, nullptr, c3Wl, DH, nullptr, h, 128, 0,  N, DH, DH);
    gemm_wmma_f32<<<gridConv, GB, 0, stream>>>(e, DH, nullptr, c3Wr, DH, nullptr, h, 128, DH, N, DH, DH);
    hipMemsetAsync(agg, 0, (size_t)N * 64 * sizeof(float), stream);
    edge_agg_kernel<<<eBlocks, TPB, 0, stream>>>(h, 128, agg, src, dst, E);
    combine_kernel<<<nBlocks16, TPB, 0, stream>>>(agg, deg, c3bl, h + DH, 128, e, nTot16);

    // ===== global mean pool =====
    hipMemsetAsync(pool, 0, (size_t)B * 64 * sizeof(float), stream);
    hipMemsetAsync(cnt, 0, (size_t)B * sizeof(float), stream);
    pool_kernel<<<nBlocks16, TPB, 0, stream>>>(e, batch, pool, nTot16);
    cnt_kernel<<<(N + TPB - 1) / TPB, TPB, 0, stream>>>(batch, cnt, N);
    pool_div_kernel<<<(B * 64 + TPB - 1) / TPB, TPB, 0, stream>>>(pool, cnt, B * 64);

    // ===== MLP head (M = B = 512, MT = 32) =====
    const int MTb = B >> 4;
    dim3 g1((MTb + 3) / 4, (HID + 15) / 16);
    gemm_wmma_f32<<<g1, GB, 0, stream>>>(pool, DH, nullptr, l1W, HID, l1b, t1, HID, 0, B, DH, HID);
    bn_tanh_kernel<<<HID, 256, 0, stream>>>(t1, B, HID, bn1g, bn1b);

    dim3 g2((MTb + 3) / 4, (H2 + 15) / 16);
    gemm_wmma_f32<<<g2, GB, 0, stream>>>(t1, HID, nullptr, l2W, H2, l2b, t2, H2, 0, B, HID, H2);
    bn_tanh_kernel<<<H2, 256, 0, stream>>>(t2, B, H2, bn2g, bn2b);

    gemm_wmma_f32<<<g2, GB, 0, stream>>>(t2, H2, nullptr, l3W, H2, l3b, t1, H2, 0, B, H2, H2);
    bn_tanh_kernel<<<H2, 256, 0, stream>>>(t1, B, H2, bn3g, bn3b);

    dim3 g4((MTb + 3) / 4, (NCL + 15) / 16);
    gemm_wmma_f32<<<g4, GB, 0, stream>>>(t1, H2, nullptr, l4W, NCL, l4b, out, NCL, 0, B, H2, NCL);
}